// GATAdapter_old_30777735643945
// MI455X (gfx1250) — compile-verified
//
#include <hip/hip_runtime.h>

// ---------------- problem constants ----------------
#define Nn    4096
#define Ee    12288
#define Bb    64
#define CLIPD 768
#define HIDD  1024
#define Hh    4
#define C2D   3072
#define EAn   (Ee + Nn)        // 16384 edges incl. self loops
#define NEG_SLOPE 0.2f

typedef float v2f __attribute__((ext_vector_type(2)));
typedef float v8f __attribute__((ext_vector_type(8)));

// ---- optional CDNA5 async global->LDS path (ASYNCcnt-tracked DMA) ----
#if defined(__has_builtin)
#  if __has_builtin(__builtin_amdgcn_global_load_async_to_lds_b128)
#    define HAVE_ASYNC_LDS 1
#  endif
#endif
#ifndef HAVE_ASYNC_LDS
#  define HAVE_ASYNC_LDS 0
#endif

#if HAVE_ASYNC_LDS
// builtin signature (from clang diagnostic): param0 = v4i in addrspace(1) (global),
// param1 = LDS side, then imm offset + cpol
typedef int v4i __attribute__((vector_size(16)));
typedef __attribute__((address_space(1))) v4i* gptr16;
typedef __attribute__((address_space(3))) v4i* lptr16;
__device__ __forceinline__ void async_cp16(const void* g, void* l) {
  // per-lane 16B copy, global -> LDS, tracked by ASYNCcnt
  __builtin_amdgcn_global_load_async_to_lds_b128((gptr16)g, (lptr16)l, 0, 0);
}
#  if __has_builtin(__builtin_amdgcn_s_wait_asynccnt)
#    define WAIT_ASYNC0() __builtin_amdgcn_s_wait_asynccnt(0)
#  else
#    define WAIT_ASYNC0() asm volatile("s_wait_asynccnt 0x0" ::: "memory")
#  endif
#endif

__device__ __forceinline__ unsigned encf(float f) {
  unsigned u = __float_as_uint(f);
  return (u & 0x80000000u) ? ~u : (u | 0x80000000u);
}
__device__ __forceinline__ float decf(unsigned u) {
  return (u & 0x80000000u) ? __uint_as_float(u ^ 0x80000000u)
                           : __uint_as_float(~u);
}

// ---------------- small reduction kernels ----------------

// w_e[d][p] : p<4 -> layer1 head p, p>=4 -> layer2 head p-4
__global__ void gat_we_kernel(const float* __restrict__ We1, const float* __restrict__ Ae1,
                              const float* __restrict__ We2, const float* __restrict__ Ae2,
                              float* __restrict__ w_e) {
  __shared__ float red[256];
  int d = blockIdx.x, t = threadIdx.x;
  for (int p = 0; p < 8; ++p) {
    const float* W; const float* A; int C;
    if (p < 4) { W = We1 + ((size_t)d * Hh + p) * HIDD; A = Ae1 + (size_t)p * HIDD; C = HIDD; }
    else       { W = We2 + ((size_t)d * Hh + (p - 4)) * C2D; A = Ae2 + (size_t)(p - 4) * C2D; C = C2D; }
    float s = 0.f;
    for (int c = t; c < C; c += 256) s += W[c] * A[c];
    red[t] = s; __syncthreads();
    for (int off = 128; off > 0; off >>= 1) { if (t < off) red[t] += red[t + off]; __syncthreads(); }
    if (t == 0) w_e[(size_t)d * 8 + p] = red[0];
    __syncthreads();
  }
}

// a_e[e][0..7] = edge_attr[e] @ w_e   (real edges only)
__global__ void gat_ae_kernel(const float* __restrict__ edge_attr,
                              const float* __restrict__ w_e,
                              float* __restrict__ a_e) {
  __shared__ float red[256];
  int e = blockIdx.x, t = threadIdx.x;
  float acc[8] = {0.f,0.f,0.f,0.f,0.f,0.f,0.f,0.f};
  for (int d = t; d < CLIPD; d += 256) {
    float v = edge_attr[(size_t)e * CLIPD + d];
    const float* w = w_e + (size_t)d * 8;
#pragma unroll
    for (int p = 0; p < 8; ++p) acc[p] += v * w[p];
  }
  for (int p = 0; p < 8; ++p) {
    red[t] = acc[p]; __syncthreads();
    for (int off = 128; off > 0; off >>= 1) { if (t < off) red[t] += red[t + off]; __syncthreads(); }
    if (t == 0) a_e[(size_t)e * 8 + p] = red[0];
    __syncthreads();
  }
}

__global__ void gat_zero_i32(int* p, int n) {
  int i = blockIdx.x * 256 + threadIdx.x;
  if (i < n) p[i] = 0;
}

__global__ void gat_deg_kernel(const int* __restrict__ ei, int* __restrict__ deg) {
  int e = blockIdx.x * 256 + threadIdx.x;
  if (e < Ee) atomicAdd(&deg[ei[Ee + e]], 1);
}

// deterministic stable rank: #edges j<e with same dst
__global__ void gat_rank_kernel(const int* __restrict__ dst, int* __restrict__ rank) {
  __shared__ int red[256];
  int e = blockIdx.x, t = threadIdx.x;
  int d = dst[e], cnt = 0;
  for (int j = t; j < e; j += 256) cnt += (dst[j] == d) ? 1 : 0;
  red[t] = cnt; __syncthreads();
  for (int off = 128; off > 0; off >>= 1) { if (t < off) red[t] += red[t + off]; __syncthreads(); }
  if (t == 0) rank[e] = red[0];
}

// exclusive scan of (deg[n]+1) over N=4096 nodes, single block of 256
__global__ void gat_scan_kernel(const int* __restrict__ deg, int* __restrict__ ptr) {
  __shared__ int part[256];
  int t = threadIdx.x, base = t * 16, s = 0;
  for (int i = 0; i < 16; ++i) s += deg[base + i] + 1;
  part[t] = s; __syncthreads();
  for (int off = 1; off < 256; off <<= 1) {
    int v = (t >= off) ? part[t - off] : 0;
    __syncthreads(); part[t] += v; __syncthreads();
  }
  int pre = (t == 0) ? 0 : part[t - 1];
  for (int i = 0; i < 16; ++i) { ptr[base + i] = pre; pre += deg[base + i] + 1; }
  if (t == 255) ptr[Nn] = pre;
}

__global__ void gat_fill_kernel(const int* __restrict__ ei, const int* __restrict__ deg,
                                const int* __restrict__ ptr, const int* __restrict__ rank,
                                int* __restrict__ eidx) {
  int e = blockIdx.x * 256 + threadIdx.x;
  if (e >= EAn) return;
  if (e < Ee) eidx[ptr[ei[Ee + e]] + rank[e]] = e;
  else        { int n = e - Ee; eidx[ptr[n] + deg[n]] = e; }
}

// self-loop edge attention = segsum(a_e)/max(deg,1), written into rows E..E+N-1
__global__ void gat_aeself_kernel(const int* __restrict__ ptr, const int* __restrict__ eidx,
                                  const int* __restrict__ deg, float* __restrict__ a_e) {
  int n = blockIdx.x * 256 + threadIdx.x;
  if (n >= Nn) return;
  float s[8] = {0.f,0.f,0.f,0.f,0.f,0.f,0.f,0.f};
  int b = ptr[n], e2 = ptr[n + 1];
  for (int j = b; j < e2; ++j) {
    int e = eidx[j];
    if (e < Ee) {
#pragma unroll
      for (int p = 0; p < 8; ++p) s[p] += a_e[(size_t)e * 8 + p];
    }
  }
  float inv = 1.f / (float)max(deg[n], 1);
#pragma unroll
  for (int p = 0; p < 8; ++p) a_e[(size_t)(Ee + n) * 8 + p] = s[p] * inv;
}

// ---------------- f32 WMMA GEMM: C[M,N] = A[M,K] @ B[K,N] ----------------
// requires M%64==0, N%128==0, K%16==0 (true for all call sites)
// mode 1: C = prelu(C + bias[col], *prelu_a)
// Per block: 64x128 tile, 8 wave32s, each wave owns a 32x32 patch (2x2 WMMA accs).
// Async path: double-buffered LDS staged via GLOBAL_LOAD_ASYNC_TO_LDS_B128,
// tile k+1 DMA overlaps the 16 v_wmma of tile k; one s_wait_asynccnt + one
// barrier per K-tile.
__global__ __launch_bounds__(256)
void gat_gemm_f32_wmma(const float* __restrict__ A, const float* __restrict__ Bm,
                       float* __restrict__ C, int M, int Ncol, int K,
                       int lda, int ldb, int ldc,
                       const float* __restrict__ bias, const float* __restrict__ prelu_a,
                       int mode) {
#if HAVE_ASYNC_LDS
  __shared__ float As[2][64][20];   // row pad 20 keeps 16B row alignment
  __shared__ float Bs[2][16][128];
#else
  __shared__ float As[64][20];
  __shared__ float Bs[16][128];
#endif

  const int t   = threadIdx.x;
  const int m0  = blockIdx.y * 64;
  const int n0  = blockIdx.x * 128;
  const int wv  = t >> 5;            // wave 0..7
  const int wm  = wv >> 2;           // 0..1 -> 32-row patch
  const int wn  = wv & 3;            // 0..3 -> 32-col patch
  const int lr  = t & 15;            // lane row/col within 16
  const int l16 = (t >> 4) & 1;      // lane half -> K pair

  // global staging mapping
  const int am = t >> 2, aq = (t & 3) * 4;       // A: 64 rows x 16 k, one b128 each
  const int kr = t >> 4, nc = (t & 15) * 8;      // B: 16 rows x 128 cols, two b128 each

  v8f acc[2][2];
  acc[0][0] = {}; acc[0][1] = {}; acc[1][0] = {}; acc[1][1] = {};

  const int lm = wm * 32, ln = wn * 32;

#if HAVE_ASYNC_LDS
  const float* agp = A  + (size_t)(m0 + am) * lda + aq;
  const float* bgp = Bm + (size_t)kr * ldb + n0 + nc;

  // prologue: tile 0 -> buffer 0
  async_cp16(agp,     &As[0][am][aq]);
  async_cp16(bgp,     &Bs[0][kr][nc]);
  async_cp16(bgp + 4, &Bs[0][kr][nc + 4]);

  int buf = 0;
  for (int k0 = 0; k0 < K; k0 += 16) {
    WAIT_ASYNC0();        // own DMA done
    __syncthreads();      // everyone's DMA done + prev compute on buf^1 done
    if (k0 + 16 < K) {    // prefetch next tile into the other buffer
      const float* bn = bgp + (size_t)(k0 + 16) * ldb;
      async_cp16(agp + k0 + 16, &As[buf ^ 1][am][aq]);
      async_cp16(bn,            &Bs[buf ^ 1][kr][nc]);
      async_cp16(bn + 4,        &Bs[buf ^ 1][kr][nc + 4]);
    }
#pragma unroll
    for (int kk = 0; kk < 16; kk += 4) {
      const int kb = kk + l16 * 2;
      v2f a0  = { As[buf][lm +  0 + lr][kb], As[buf][lm +  0 + lr][kb + 1] };
      v2f a1  = { As[buf][lm + 16 + lr][kb], As[buf][lm + 16 + lr][kb + 1] };
      v2f b0f = { Bs[buf][kb][ln +  0 + lr], Bs[buf][kb + 1][ln +  0 + lr] };
      v2f b1f = { Bs[buf][kb][ln + 16 + lr], Bs[buf][kb + 1][ln + 16 + lr] };
      acc[0][0] = __builtin_amdgcn_wmma_f32_16x16x4_f32(false, a0, false, b0f, (short)0, acc[0][0], false, false);
      acc[0][1] = __builtin_amdgcn_wmma_f32_16x16x4_f32(false, a0, false, b1f, (short)0, acc[0][1], false, false);
      acc[1][0] = __builtin_amdgcn_wmma_f32_16x16x4_f32(false, a1, false, b0f, (short)0, acc[1][0], false, false);
      acc[1][1] = __builtin_amdgcn_wmma_f32_16x16x4_f32(false, a1, false, b1f, (short)0, acc[1][1], false, false);
    }
    buf ^= 1;
  }
#else
  for (int k0 = 0; k0 < K; k0 += 16) {
    float4 av = *(const float4*)(A + (size_t)(m0 + am) * lda + k0 + aq);
    As[am][aq + 0] = av.x; As[am][aq + 1] = av.y; As[am][aq + 2] = av.z; As[am][aq + 3] = av.w;
    const float4* bsrc = (const float4*)(Bm + (size_t)(k0 + kr) * ldb + n0 + nc);
    float4 b0 = bsrc[0], b1 = bsrc[1];
    ((float4*)&Bs[kr][nc])[0] = b0;
    ((float4*)&Bs[kr][nc])[1] = b1;
    if (k0 + 16 < K) {  // hint next K-tile (global_prefetch_b8)
      __builtin_prefetch(A + (size_t)(m0 + am) * lda + (k0 + 16) + aq, 0, 0);
      __builtin_prefetch(Bm + (size_t)(k0 + 16 + kr) * ldb + n0 + nc, 0, 0);
    }
    __syncthreads();

#pragma unroll
    for (int kk = 0; kk < 16; kk += 4) {
      const int kb = kk + l16 * 2;
      v2f a0  = { As[lm +  0 + lr][kb], As[lm +  0 + lr][kb + 1] };
      v2f a1  = { As[lm + 16 + lr][kb], As[lm + 16 + lr][kb + 1] };
      v2f b0f = { Bs[kb][ln +  0 + lr], Bs[kb + 1][ln +  0 + lr] };
      v2f b1f = { Bs[kb][ln + 16 + lr], Bs[kb + 1][ln + 16 + lr] };
      acc[0][0] = __builtin_amdgcn_wmma_f32_16x16x4_f32(false, a0, false, b0f, (short)0, acc[0][0], false, false);
      acc[0][1] = __builtin_amdgcn_wmma_f32_16x16x4_f32(false, a0, false, b1f, (short)0, acc[0][1], false, false);
      acc[1][0] = __builtin_amdgcn_wmma_f32_16x16x4_f32(false, a1, false, b0f, (short)0, acc[1][0], false, false);
      acc[1][1] = __builtin_amdgcn_wmma_f32_16x16x4_f32(false, a1, false, b1f, (short)0, acc[1][1], false, false);
    }
    __syncthreads();
  }
#endif

  const float pa = (mode == 1) ? prelu_a[0] : 0.f;
#pragma unroll
  for (int i = 0; i < 2; ++i)
#pragma unroll
    for (int j = 0; j < 2; ++j) {
      const int colg = n0 + ln + j * 16 + lr;
      const float bv = (mode == 1) ? bias[colg] : 0.f;
#pragma unroll
      for (int r = 0; r < 8; ++r) {
        const int rowg = m0 + lm + i * 16 + l16 * 8 + r;
        float v = acc[i][j][r];
        if (mode == 1) { v += bv; v = (v >= 0.f) ? v : pa * v; }
        C[(size_t)rowg * ldc + colg] = v;
      }
    }
}

// ---------------- attention scores / softmax ----------------

// asd[n][h] = <xl[n,h,:], att_src[h,:]> ; asd[n][4+h] = <xl[n,h,:], att_dst[h,:]>
__global__ void gat_asd_kernel(const float* __restrict__ xl,
                               const float* __restrict__ att_s, const float* __restrict__ att_d,
                               int C, float* __restrict__ asd) {
  __shared__ float red[256];
  int n = blockIdx.x, t = threadIdx.x;
  for (int h = 0; h < Hh; ++h) {
    const float* row = xl + ((size_t)n * Hh + h) * C;
    const float* as = att_s + (size_t)h * C;
    const float* ad = att_d + (size_t)h * C;
    float s = 0.f, dd = 0.f;
    for (int c = t; c < C; c += 256) { float v = row[c]; s += v * as[c]; dd += v * ad[c]; }
    red[t] = s; __syncthreads();
    for (int off = 128; off > 0; off >>= 1) { if (t < off) red[t] += red[t + off]; __syncthreads(); }
    if (t == 0) asd[(size_t)n * 8 + h] = red[0];
    __syncthreads();
    red[t] = dd; __syncthreads();
    for (int off = 128; off > 0; off >>= 1) { if (t < off) red[t] += red[t + off]; __syncthreads(); }
    if (t == 0) asd[(size_t)n * 8 + 4 + h] = red[0];
    __syncthreads();
  }
}

__global__ void gat_init_amax(unsigned* amax, int n) {
  int i = blockIdx.x * 256 + threadIdx.x;
  if (i < n) amax[i] = 0x007FFFFFu;   // enc(-inf)
}

__global__ void gat_alpha_kernel(const int* __restrict__ ei, const float* __restrict__ asd,
                                 const float* __restrict__ a_e, int aeOff,
                                 float* __restrict__ ex, unsigned* __restrict__ amax) {
  int e = blockIdx.x * 256 + threadIdx.x;
  if (e >= EAn) return;
  int sn, dn;
  if (e < Ee) { sn = ei[e]; dn = ei[Ee + e]; } else { sn = dn = e - Ee; }
#pragma unroll
  for (int h = 0; h < Hh; ++h) {
    float al = asd[(size_t)sn * 8 + h] + asd[(size_t)dn * 8 + 4 + h]
             + a_e[(size_t)e * 8 + aeOff + h];
    al = (al >= 0.f) ? al : NEG_SLOPE * al;          // leaky_relu
    ex[(size_t)e * Hh + h] = al;
    atomicMax(&amax[(size_t)dn * Hh + h], encf(al));
  }
}

__global__ void gat_exp_kernel(const int* __restrict__ ei, float* __restrict__ ex,
                               const unsigned* __restrict__ amax) {
  int e = blockIdx.x * 256 + threadIdx.x;
  if (e >= EAn) return;
  int dn = (e < Ee) ? ei[Ee + e] : (e - Ee);
#pragma unroll
  for (int h = 0; h < Hh; ++h) {
    float m = decf(amax[(size_t)dn * Hh + h]);
    ex[(size_t)e * Hh + h] = expf(ex[(size_t)e * Hh + h] - m);
  }
}

// deterministic per-destination denominator (CSR order)
__global__ void gat_den_kernel(const int* __restrict__ ptr, const int* __restrict__ eidx,
                               const float* __restrict__ ex, float* __restrict__ den) {
  int n = blockIdx.x * 256 + threadIdx.x;
  if (n >= Nn) return;
  int b = ptr[n], e2 = ptr[n + 1];
  float s[Hh] = {0.f,0.f,0.f,0.f};
  for (int j = b; j < e2; ++j) {
    int e = eidx[j];
#pragma unroll
    for (int h = 0; h < Hh; ++h) s[h] += ex[(size_t)e * Hh + h];
  }
#pragma unroll
  for (int h = 0; h < Hh; ++h) den[(size_t)n * Hh + h] = s[h];
}

// fused aggregation + head-mean + bias + PReLU  -> hout[N,C]
__global__ void gat_agg_kernel(const float* __restrict__ xl, const float* __restrict__ ex,
                               const float* __restrict__ den, const int* __restrict__ eidx,
                               const int* __restrict__ ptr, const int* __restrict__ ei,
                               const float* __restrict__ bias, const float* __restrict__ prelu_a,
                               int C, float* __restrict__ hout) {
  int n = blockIdx.y;
  int c = blockIdx.x * 256 + threadIdx.x;
  if (c >= C) return;
  float invd[Hh];
#pragma unroll
  for (int h = 0; h < Hh; ++h) invd[h] = 1.f / (den[(size_t)n * Hh + h] + 1e-16f);
  int b = ptr[n], e2 = ptr[n + 1];
  float acc = 0.f;
  for (int j = b; j < e2; ++j) {
    int e = eidx[j];
    int sn = (e < Ee) ? ei[e] : (e - Ee);
    const float* xr = xl + (size_t)sn * Hh * C + c;
#pragma unroll
    for (int h = 0; h < Hh; ++h)
      acc += ex[(size_t)e * Hh + h] * invd[h] * xr[(size_t)h * C];
  }
  float a = prelu_a[0];
  float v = acc * 0.25f + bias[c];
  hout[(size_t)n * C + c] = (v >= 0.f) ? v : a * v;
}

// ---------------- gate + attentional pooling ----------------

__global__ void gat_gate_kernel(const float* __restrict__ g2, const float* __restrict__ gW3,
                                const float* __restrict__ gb3, float* __restrict__ gate) {
  __shared__ float red[256];
  int n = blockIdx.x, t = threadIdx.x;
  float s = 0.f;
  for (int c = t; c < HIDD; c += 256) s += g2[(size_t)n * HIDD + c] * gW3[c];
  red[t] = s; __syncthreads();
  for (int off = 128; off > 0; off >>= 1) { if (t < off) red[t] += red[t + off]; __syncthreads(); }
  if (t == 0) gate[n] = red[0] + gb3[0];
}

__global__ void gat_bptr_kernel(const int* __restrict__ batch, int* __restrict__ bptr) {
  int b = threadIdx.x;
  if (b > Bb) return;
  int lo = 0, hi = Nn;
  while (lo < hi) { int mid = (lo + hi) >> 1; if (batch[mid] < b) lo = mid + 1; else hi = mid; }
  bptr[b] = lo;
}

__global__ void gat_gmax_kernel(const int* __restrict__ bptr, const float* __restrict__ gate,
                                float* __restrict__ gmax) {
  int b = threadIdx.x;
  if (b >= Bb) return;
  int s = bptr[b], e = bptr[b + 1];
  float m = -__builtin_inff();
  for (int n = s; n < e; ++n) m = fmaxf(m, gate[n]);
  if (e == s || !isfinite(m)) m = 0.f;
  gmax[b] = m;
}

__global__ void gat_eg_kernel(const float* __restrict__ gate, const float* __restrict__ gmax,
                              const int* __restrict__ batch, float* __restrict__ eg) {
  int n = blockIdx.x * 256 + threadIdx.x;
  if (n < Nn) eg[n] = expf(gate[n] - gmax[batch[n]]);
}

__global__ void gat_gden_kernel(const int* __restrict__ bptr, const float* __restrict__ eg,
                                float* __restrict__ gden) {
  int b = threadIdx.x;
  if (b >= Bb) return;
  int s = bptr[b], e = bptr[b + 1];
  float d = 0.f;
  for (int n = s; n < e; ++n) d += eg[n];
  gden[b] = d;
}

__global__ void gat_pool_kernel(const int* __restrict__ bptr, const float* __restrict__ eg,
                                const float* __restrict__ gden, const float* __restrict__ h2,
                                float* __restrict__ out) {
  int b = blockIdx.y;
  int c = blockIdx.x * 256 + threadIdx.x;
  int s = bptr[b], e = bptr[b + 1];
  float acc = 0.f;
  for (int n = s; n < e; ++n) acc += eg[n] * h2[(size_t)n * C2D + c];
  out[(size_t)b * C2D + c] = acc / (gden[b] + 1e-16f);
}

// ---------------- host driver ----------------
extern "C" void kernel_launch(void* const* d_in, const int* in_sizes, int n_in,
                              void* d_out, int out_size, void* d_ws, size_t ws_size,
                              hipStream_t stream) {
  const float* x         = (const float*)d_in[0];
  const int*   ei        = (const int*)  d_in[1];   // [2,E] row-major: src=ei[e], dst=ei[E+e]
  const float* edge_attr = (const float*)d_in[2];
  const int*   batch     = (const int*)  d_in[3];
  const float* W1        = (const float*)d_in[4];
  const float* att_src1  = (const float*)d_in[5];
  const float* att_dst1  = (const float*)d_in[6];
  const float* W_edge1   = (const float*)d_in[7];
  const float* att_edge1 = (const float*)d_in[8];
  const float* bias1     = (const float*)d_in[9];
  const float* prelu1    = (const float*)d_in[10];
  const float* W2        = (const float*)d_in[11];
  const float* att_src2  = (const float*)d_in[12];
  const float* att_dst2  = (const float*)d_in[13];
  const float* W_edge2   = (const float*)d_in[14];
  const float* att_edge2 = (const float*)d_in[15];
  const float* bias2     = (const float*)d_in[16];
  const float* prelu2    = (const float*)d_in[17];
  const float* gW1       = (const float*)d_in[18];
  const float* gb1       = (const float*)d_in[19];
  const float* ga1       = (const float*)d_in[20];
  const float* gW2       = (const float*)d_in[21];
  const float* gb2       = (const float*)d_in[22];
  const float* ga2       = (const float*)d_in[23];
  const float* gW3       = (const float*)d_in[24];
  const float* gb3       = (const float*)d_in[25];
  float* out = (float*)d_out;

  // workspace carve
  char* base = (char*)d_ws;
  size_t off = 0;
  auto carve = [&](size_t bytes) -> void* {
    off = (off + 255) & ~(size_t)255;
    void* p = base + off;
    off += bytes;
    return p;
  };
  float*    xl    = (float*)carve((size_t)Nn * Hh * C2D * 4);  // reused by both layers
  float*    h1    = (float*)carve((size_t)Nn * HIDD * 4);
  float*    h2    = (float*)carve((size_t)Nn * C2D * 4);
  float*    g1    = (float*)carve((size_t)Nn * HIDD * 4);
  float*    g2    = (float*)carve((size_t)Nn * HIDD * 4);
  float*    w_e   = (float*)carve((size_t)CLIPD * 8 * 4);
  float*    a_e   = (float*)carve((size_t)EAn * 8 * 4);
  float*    asd   = (float*)carve((size_t)Nn * 8 * 4);
  int*      deg   = (int*)  carve((size_t)Nn * 4);
  int*      rank  = (int*)  carve((size_t)Ee * 4);
  int*      ptr   = (int*)  carve((size_t)(Nn + 1) * 4);
  int*      eidx  = (int*)  carve((size_t)EAn * 4);
  unsigned* amax  = (unsigned*)carve((size_t)Nn * Hh * 4);
  float*    den   = (float*)carve((size_t)Nn * Hh * 4);
  float*    ex    = (float*)carve((size_t)EAn * Hh * 4);
  float*    gate  = (float*)carve((size_t)Nn * 4);
  float*    eg    = (float*)carve((size_t)Nn * 4);
  int*      bptr  = (int*)  carve((size_t)(Bb + 1) * 4);
  float*    gmax  = (float*)carve((size_t)Bb * 4);
  float*    gden  = (float*)carve((size_t)Bb * 4);
  (void)ws_size; (void)in_sizes; (void)n_in; (void)out_size;

  // ---- stage 0: fused edge attention + deterministic CSR ----
  gat_we_kernel<<<CLIPD, 256, 0, stream>>>(W_edge1, att_edge1, W_edge2, att_edge2, w_e);
  gat_ae_kernel<<<Ee, 256, 0, stream>>>(edge_attr, w_e, a_e);
  gat_zero_i32<<<(Nn + 255) / 256, 256, 0, stream>>>(deg, Nn);
  gat_deg_kernel<<<(Ee + 255) / 256, 256, 0, stream>>>(ei, deg);
  gat_rank_kernel<<<Ee, 256, 0, stream>>>(ei + Ee, rank);
  gat_scan_kernel<<<1, 256, 0, stream>>>(deg, ptr);
  gat_fill_kernel<<<(EAn + 255) / 256, 256, 0, stream>>>(ei, deg, ptr, rank, eidx);
  gat_aeself_kernel<<<(Nn + 255) / 256, 256, 0, stream>>>(ptr, eidx, deg, a_e);

  // ---- GATConv layer 1 ----
  gat_gemm_f32_wmma<<<dim3((Hh * HIDD) / 128, Nn / 64), 256, 0, stream>>>(
      x, W1, xl, Nn, Hh * HIDD, CLIPD, CLIPD, Hh * HIDD, Hh * HIDD, nullptr, nullptr, 0);
  gat_asd_kernel<<<Nn, 256, 0, stream>>>(xl, att_src1, att_dst1, HIDD, asd);
  gat_init_amax<<<(Nn * Hh + 255) / 256, 256, 0, stream>>>(amax, Nn * Hh);
  gat_alpha_kernel<<<(EAn + 255) / 256, 256, 0, stream>>>(ei, asd, a_e, 0, ex, amax);
  gat_exp_kernel<<<(EAn + 255) / 256, 256, 0, stream>>>(ei, ex, amax);
  gat_den_kernel<<<(Nn + 255) / 256, 256, 0, stream>>>(ptr, eidx, ex, den);
  gat_agg_kernel<<<dim3(HIDD / 256, Nn), 256, 0, stream>>>(
      xl, ex, den, eidx, ptr, ei, bias1, prelu1, HIDD, h1);

  // ---- GATConv layer 2 ----
  gat_gemm_f32_wmma<<<dim3((Hh * C2D) / 128, Nn / 64), 256, 0, stream>>>(
      h1, W2, xl, Nn, Hh * C2D, HIDD, HIDD, Hh * C2D, Hh * C2D, nullptr, nullptr, 0);
  gat_asd_kernel<<<Nn, 256, 0, stream>>>(xl, att_src2, att_dst2, C2D, asd);
  gat_init_amax<<<(Nn * Hh + 255) / 256, 256, 0, stream>>>(amax, Nn * Hh);
  gat_alpha_kernel<<<(EAn + 255) / 256, 256, 0, stream>>>(ei, asd, a_e, 4, ex, amax);
  gat_exp_kernel<<<(EAn + 255) / 256, 256, 0, stream>>>(ei, ex, amax);
  gat_den_kernel<<<(Nn + 255) / 256, 256, 0, stream>>>(ptr, eidx, ex, den);
  gat_agg_kernel<<<dim3(C2D / 256, Nn), 256, 0, stream>>>(
      xl, ex, den, eidx, ptr, ei, bias2, prelu2, C2D, h2);

  // ---- gate MLP (fused bias+PReLU epilogues) ----
  gat_gemm_f32_wmma<<<dim3(HIDD / 128, Nn / 64), 256, 0, stream>>>(
      h2, gW1, g1, Nn, HIDD, C2D, C2D, HIDD, HIDD, gb1, ga1, 1);
  gat_gemm_f32_wmma<<<dim3(HIDD / 128, Nn / 64), 256, 0, stream>>>(
      g1, gW2, g2, Nn, HIDD, HIDD, HIDD, HIDD, HIDD, gb2, ga2, 1);
  gat_gate_kernel<<<Nn, 256, 0, stream>>>(g2, gW3, gb3, gate);

  // ---- attentional pooling over batch segments ----
  gat_bptr_kernel<<<1, 128, 0, stream>>>(batch, bptr);
  gat_gmax_kernel<<<1, Bb, 0, stream>>>(bptr, gate, gmax);
  gat_eg_kernel<<<(Nn + 255) / 256, 256, 0, stream>>>(gate, gmax, batch, eg);
  gat_gden_kernel<<<1, Bb, 0, stream>>>(bptr, eg, gden);
  gat_pool_kernel<<<dim3(C2D / 256, Bb), 256, 0, stream>>>(bptr, eg, gden, h2, out);
}